// MHGCN_21801253994613
// MI455X (gfx1250) — compile-verified
//
#include <hip/hip_runtime.h>

typedef __attribute__((ext_vector_type(2))) float v2f;
typedef __attribute__((ext_vector_type(8))) float v8f;

constexpr int NN = 8192;       // graph size
constexpr int F  = 64;         // feature / output width
constexpr int KC = 64;         // K chunk staged in LDS
constexpr int RT = 64;         // row tile per block
constexpr int KSPLIT = 4;      // K partitions (grid.y)
constexpr int KLEN = NN / KSPLIT;

constexpr int ASTR = KC + 4;   // 68: As stride  -> banks 4*row+k, conflict-free
constexpr int TSTR = RT + 8;   // 72: At stride  -> K-halves 16 banks apart
constexpr int BSTR = F  + 8;   // 72: Bs stride  -> K-halves 16 banks apart

// ---------------------------------------------------------------------------
// Kernel 1: temp[i,j] = sum_k A[i,j,k] * w[k]   (A is [N,N,5], innermost k)
// Each thread produces 4 outputs = 20 input floats = 5 aligned float4 loads.
// ---------------------------------------------------------------------------
__global__ __launch_bounds__(256) void k_temp(const float* __restrict__ A,
                                              const float* __restrict__ wb,
                                              float* __restrict__ temp) {
  long long g = (long long)blockIdx.x * 256 + threadIdx.x;  // 0 .. N*N/4-1
  float w0 = wb[0], w1 = wb[1], w2 = wb[2], w3 = wb[3], w4 = wb[4];
  union { float4 v[5]; float f[20]; } u;
  const float4* p = (const float4*)(A + 20ll * g);  // 80B-aligned
#pragma unroll
  for (int i = 0; i < 5; ++i) u.v[i] = p[i];
  float4 o;
  o.x = u.f[0]  * w0 + u.f[1]  * w1 + u.f[2]  * w2 + u.f[3]  * w3 + u.f[4]  * w4;
  o.y = u.f[5]  * w0 + u.f[6]  * w1 + u.f[7]  * w2 + u.f[8]  * w3 + u.f[9]  * w4;
  o.z = u.f[10] * w0 + u.f[11] * w1 + u.f[12] * w2 + u.f[13] * w3 + u.f[14] * w4;
  o.w = u.f[15] * w0 + u.f[16] * w1 + u.f[17] * w2 + u.f[18] * w3 + u.f[19] * w4;
  ((float4*)temp)[g] = o;
}

// ---------------------------------------------------------------------------
// Kernel 2: X[8192x64] = Ain[8192x64] @ W[64x64]   (tiny)
// ---------------------------------------------------------------------------
__global__ __launch_bounds__(256) void k_xw(const float* __restrict__ Ain,
                                            const float* __restrict__ W,
                                            float* __restrict__ X) {
  __shared__ float Ws[64 * 64];
  int t = threadIdx.x;
  float4 wr4[4];
#pragma unroll
  for (int i = 0; i < 4; ++i) wr4[i] = ((const float4*)W)[t + 256 * i];
#pragma unroll
  for (int i = 0; i < 4; ++i) ((float4*)Ws)[t + 256 * i] = wr4[i];
  __syncthreads();
  int row = blockIdx.x * 4 + (t >> 6);
  int col = t & 63;
  const float4* a4 = (const float4*)(Ain + row * F);
  float s = 0.f;
#pragma unroll
  for (int k4 = 0; k4 < 16; ++k4) {
    float4 av = a4[k4];
    s += av.x * Ws[(4 * k4 + 0) * 64 + col] + av.y * Ws[(4 * k4 + 1) * 64 + col] +
         av.z * Ws[(4 * k4 + 2) * 64 + col] + av.w * Ws[(4 * k4 + 3) * 64 + col];
  }
  X[row * F + col] = s;
}

// ---------------------------------------------------------------------------
// Kernel 3: partial[ks] = (temp + temp^T)[:, Krange(ks)] @ X[Krange(ks), :]
// fp32 WMMA (V_WMMA_F32_16X16X4_F32). Block = 256 thr = 8 waves.
// Block tile 64x64; wave tile 16x32 (2 accumulators). Register double-buffered
// LDS staging with float4 global loads. Symmetrization folded into A-fragment.
// ---------------------------------------------------------------------------
__global__ __launch_bounds__(256) void k_spmm(const float* __restrict__ temp,
                                              const float* __restrict__ X,
                                              float* __restrict__ partial) {
  __shared__ float As[RT * ASTR];   // row slab
  __shared__ float At[KC * TSTR];   // transposed slab (stored k-major)
  __shared__ float Bs[KC * BSTR];

  const int t     = threadIdx.x;
  const int row0  = blockIdx.x * RT;
  const int kbase = blockIdx.y * KLEN;
  const int w     = t >> 5;
  const int lane  = t & 31;
  const int wr    = (w & 3) * 16;   // wave row offset
  const int wc    = (w >> 2) * 32;  // wave col offset
  const int mrow  = lane & 15;
  const int kh    = lane >> 4;

  float4 ra[4], rt_[4], rb[4];      // next-chunk staging registers

  auto load_chunk = [&](int kc) {
#pragma unroll
    for (int i = 0; i < 4; ++i) {   // row slab: 64 rows x 16 float4
      int idx = t + 256 * i;
      int r = idx >> 4, c4 = idx & 15;
      ra[i] = *(const float4*)(temp + (long long)(row0 + r) * NN + kc + 4 * c4);
    }
#pragma unroll
    for (int i = 0; i < 4; ++i) {   // transposed slab: 64 k-rows x 16 float4
      int idx = t + 256 * i;
      int c = idx >> 4, r4 = idx & 15;
      rt_[i] = *(const float4*)(temp + (long long)(kc + c) * NN + row0 + 4 * r4);
    }
#pragma unroll
    for (int i = 0; i < 4; ++i) {   // B tile
      int idx = t + 256 * i;
      int k = idx >> 4, n4 = idx & 15;
      rb[i] = *(const float4*)(X + (kc + k) * F + 4 * n4);
    }
  };

  auto store_chunk = [&]() {
#pragma unroll
    for (int i = 0; i < 4; ++i) {
      int idx = t + 256 * i;
      int r = idx >> 4, c4 = idx & 15;
      *(float4*)&As[r * ASTR + 4 * c4] = ra[i];
    }
#pragma unroll
    for (int i = 0; i < 4; ++i) {
      int idx = t + 256 * i;
      int c = idx >> 4, r4 = idx & 15;
      *(float4*)&At[c * TSTR + 4 * r4] = rt_[i];
    }
#pragma unroll
    for (int i = 0; i < 4; ++i) {
      int idx = t + 256 * i;
      int k = idx >> 4, n4 = idx & 15;
      *(float4*)&Bs[k * BSTR + 4 * n4] = rb[i];
    }
  };

  v8f acc0 = {};
  v8f acc1 = {};

  load_chunk(kbase);
  const int arow_ = (wr + mrow) * ASTR;
  const int trow_ = wr + mrow;

  for (int kc = kbase; kc < kbase + KLEN; kc += KC) {
    __syncthreads();                 // previous chunk's compute done with LDS
    store_chunk();
    __syncthreads();
    if (kc + KC < kbase + KLEN) load_chunk(kc + KC);  // overlap with WMMA below

#pragma unroll
    for (int k0 = 0; k0 < KC; k0 += 4) {
      int ka = k0 + kh * 2;
      v2f a;                         // final_A fragment = row slab + transposed
      a.x = As[arow_ + ka]     + At[ka * TSTR + trow_];
      a.y = As[arow_ + ka + 1] + At[(ka + 1) * TSTR + trow_];
      v2f b0;
      b0.x = Bs[ka * BSTR + wc + mrow];
      b0.y = Bs[(ka + 1) * BSTR + wc + mrow];
      v2f b1;
      b1.x = Bs[ka * BSTR + wc + 16 + mrow];
      b1.y = Bs[(ka + 1) * BSTR + wc + 16 + mrow];
      acc0 = __builtin_amdgcn_wmma_f32_16x16x4_f32(false, a, false, b0,
                                                   (short)0, acc0, false, false);
      acc1 = __builtin_amdgcn_wmma_f32_16x16x4_f32(false, a, false, b1,
                                                   (short)0, acc1, false, false);
    }
  }

  // Raw partial store (C layout: VGPR v, lane L -> M = v + 8*(L>>4), N = L&15)
  float* P = partial + (size_t)blockIdx.y * NN * F;
#pragma unroll
  for (int v = 0; v < 8; ++v) {
    int gr = row0 + wr + v + 8 * kh;
    P[gr * F + wc + mrow]      = acc0[v];
    P[gr * F + wc + 16 + mrow] = acc1[v];
  }
}

// ---------------------------------------------------------------------------
// Kernel 4: combine K-split partials + bias (+ final 0.5*(U1+U2) epilogue)
// ---------------------------------------------------------------------------
__global__ __launch_bounds__(256) void k_combine(const float* __restrict__ partial,
                                                 const float* __restrict__ bias,
                                                 const float* __restrict__ prev,
                                                 float* __restrict__ out, int mode) {
  int i = blockIdx.x * 256 + threadIdx.x;  // 0 .. NN*F-1
  int col = i & 63;
  float s = bias[col];
#pragma unroll
  for (int p = 0; p < KSPLIT; ++p) s += partial[(size_t)p * NN * F + i];
  out[i] = (mode == 0) ? s : 0.5f * (s + prev[i]);
}

// ---------------------------------------------------------------------------
// Launch: inputs A_stack, feature, weight_b, W1, b1, W2, b2
// ws: temp (268MB) | X1 | U1 | X2 (2MB each) | partial (8MB)
// ---------------------------------------------------------------------------
extern "C" void kernel_launch(void* const* d_in, const int* in_sizes, int n_in,
                              void* d_out, int out_size, void* d_ws, size_t ws_size,
                              hipStream_t stream) {
  const float* A_stack  = (const float*)d_in[0];
  const float* feature  = (const float*)d_in[1];
  const float* weight_b = (const float*)d_in[2];
  const float* W1       = (const float*)d_in[3];
  const float* b1       = (const float*)d_in[4];
  const float* W2       = (const float*)d_in[5];
  const float* b2       = (const float*)d_in[6];
  float* out = (float*)d_out;

  char* ws    = (char*)d_ws;
  float* temp = (float*)ws;                                    // NN*NN
  float* X1   = (float*)(ws + (size_t)NN * NN * sizeof(float));
  float* U1   = X1 + (size_t)NN * F;
  float* X2   = U1 + (size_t)NN * F;
  float* part = X2 + (size_t)NN * F;                           // KSPLIT*NN*F

  // 1) temp = A_stack ·k w
  k_temp<<<(NN * (long long)NN) / 4 / 256, 256, 0, stream>>>(A_stack, weight_b, temp);
  // 2) X1 = feature @ W1
  k_xw<<<NN / 4, 256, 0, stream>>>(feature, W1, X1);
  // 3) U1 = (temp+temp^T) @ X1 + b1   (K-split + combine)
  k_spmm<<<dim3(NN / RT, KSPLIT), 256, 0, stream>>>(temp, X1, part);
  k_combine<<<NN * F / 256, 256, 0, stream>>>(part, b1, U1, U1, 0);
  // 4) X2 = U1 @ W2
  k_xw<<<NN / 4, 256, 0, stream>>>(U1, W2, X2);
  // 5) out = 0.5*((temp+temp^T) @ X2 + b2 + U1)
  k_spmm<<<dim3(NN / RT, KSPLIT), 256, 0, stream>>>(temp, X2, part);
  k_combine<<<NN * F / 256, 256, 0, stream>>>(part, b2, U1, out, 1);
}